// GE2E_86363202388474
// MI455X (gfx1250) — compile-verified
//
#include <hip/hip_runtime.h>
#include <hip/hip_bf16.h>
#include <math.h>

// GE2E loss: N=1024 speakers, M=16 utterances, D=512.
// Heavy op: 16 independent 1024x1024x512 fp32 GEMMs -> V_WMMA_F32_16X16X4_F32,
// with all LDS staging done via GLOBAL_LOAD_ASYNC_TO_LDS (ASYNCcnt) and a
// double-buffered B panel so copies overlap WMMA.

#define N_SPKS 1024
#define N_UTTS 16
#define DIM    512

typedef __attribute__((ext_vector_type(2))) float v2f;
typedef __attribute__((ext_vector_type(8))) float v8f;
typedef __attribute__((ext_vector_type(2))) int   v2i;

// ---- async copy-to-LDS plumbing -------------------------------------------
// Probe-confirmed signature (from hipcc diagnostic):
//   __builtin_amdgcn_global_load_async_to_lds_b64(v2i AS1* gsrc, v2i AS3* lds,
//                                                 imm int offset, imm int cpol)
#if defined(__HIP_DEVICE_COMPILE__) && __has_builtin(__builtin_amdgcn_global_load_async_to_lds_b64)
#define GE2E_ASYNC_LDS 1
typedef __attribute__((address_space(1))) v2i* g_v2i_ptr;
typedef __attribute__((address_space(3))) v2i* l_v2i_ptr;
#define ASYNC_CP_B64(gsrc, ldst)                                               \
    __builtin_amdgcn_global_load_async_to_lds_b64((g_v2i_ptr)(gsrc),           \
                                                  (l_v2i_ptr)(ldst), 0, 0)
__device__ __forceinline__ void ge2e_wait_async() {
#if __has_builtin(__builtin_amdgcn_s_wait_asynccnt)
    __builtin_amdgcn_s_wait_asynccnt(0);
#else
    asm volatile("s_wait_asynccnt 0x0" ::: "memory");
#endif
}
#else
#define GE2E_ASYNC_LDS 0
__device__ __forceinline__ void ge2e_wait_async() {}
#endif

// ---------------------------------------------------------------------------
// Kernel 0: zero the scalar output (harness poisons d_out).
// ---------------------------------------------------------------------------
__global__ void ge2e_zero_out(float* out) {
    if (threadIdx.x == 0) out[0] = 0.0f;
}

// ---------------------------------------------------------------------------
// Kernel 1: per-speaker preprocessing.
// One block per speaker s (1024 blocks, 256 threads).
// Produces, for each u: e_n[u][s][:] and c_n[u][s][:] (L2-normalized rows),
// laid out as 16 row-major 1024x512 matrices each.
// ---------------------------------------------------------------------------
__global__ __launch_bounds__(256) void ge2e_prep(const float* __restrict__ emb,
                                                 float* __restrict__ e_n,
                                                 float* __restrict__ c_n) {
    __shared__ float se[N_UTTS][DIM];   // 32 KB
    __shared__ float ssum[DIM];         // 2 KB
    __shared__ float2 red[256];

    const int s   = blockIdx.x;
    const int tid = threadIdx.x;
    const float* base = emb + (size_t)s * N_UTTS * DIM;

    // Load 16x512 floats; batch 16 float2 loads per thread before storing so
    // the loads pipeline instead of load->wait->store chains.
    {
        float2 t[16];
        #pragma unroll
        for (int j = 0; j < 16; ++j) t[j] = ((const float2*)base)[tid + j * 256];
        #pragma unroll
        for (int j = 0; j < 16; ++j) {
            int i = tid + j * 256;
            int r = i >> 8;
            int c = (i & 255) << 1;
            se[r][c]     = t[j].x;
            se[r][c + 1] = t[j].y;
        }
    }
    __syncthreads();

    for (int d = tid; d < DIM; d += 256) {
        float acc = 0.0f;
        #pragma unroll
        for (int r = 0; r < N_UTTS; ++r) acc += se[r][d];
        ssum[d] = acc;
    }
    __syncthreads();

    const float inv_m1 = 1.0f / (float)(N_UTTS - 1);
    for (int u = 0; u < N_UTTS; ++u) {
        float pe = 0.0f, pc = 0.0f;
        for (int d = tid; d < DIM; d += 256) {
            float e = se[u][d];
            float c = (ssum[d] - e) * inv_m1;
            pe += e * e;
            pc += c * c;
        }
        red[tid].x = pe;
        red[tid].y = pc;
        __syncthreads();
        for (int off = 128; off > 0; off >>= 1) {
            if (tid < off) {
                red[tid].x += red[tid + off].x;
                red[tid].y += red[tid + off].y;
            }
            __syncthreads();
        }
        const float inv_e = 1.0f / sqrtf(red[0].x);
        const float inv_c = 1.0f / sqrtf(red[0].y);
        __syncthreads();

        float* eo = e_n + ((size_t)u * N_SPKS + s) * DIM;
        float* co = c_n + ((size_t)u * N_SPKS + s) * DIM;
        for (int d = tid; d < DIM; d += 256) {
            float e = se[u][d];
            float c = (ssum[d] - e) * inv_m1;
            eo[d] = e * inv_e;
            co[d] = c * inv_c;
        }
        __syncthreads();
    }
}

// ---------------------------------------------------------------------------
// Kernel 2: fused score GEMM (fp32 WMMA) + online log-softmax + diagonal pick.
// Grid: (64 s-tiles, 16 u). Block: 256 threads = 8 wave32.
//   score[s][t] = sum_d E_u[s][d] * C_u[t][d]   (A = E tile, B = C^T tile)
// K=512 consumed as 32 chunks of 16, double-buffered in LDS via async copies.
// LDS: 32,896 (E) + 2*9,216 (Cc ping/pong) + 8,256 (S) + 64 = 59,648 B.
// ---------------------------------------------------------------------------
#define E_STRIDE 514        // 512 + 2 pad -> conflict-free b64 operand reads
#define KC 16               // K-chunk staged in LDS
#define C_STRIDE 18         // 16 + 2 pad
#define S_STRIDE 129        // 128 + 1 pad

// Stage one 128-row x 16-float K-chunk of C into an LDS buffer.
__device__ __forceinline__ void ge2e_stage_chunk(const float* __restrict__ gsrc,
                                                 float* __restrict__ buf, int tid) {
#if GE2E_ASYNC_LDS
    #pragma unroll
    for (int j = 0; j < 4; ++j) {
        int i  = tid + j * 256;   // float2 index over 1024 (8 per row)
        int r  = i >> 3;
        int c2 = i & 7;
        ASYNC_CP_B64(gsrc + (size_t)r * DIM + (c2 << 1),
                     buf + r * C_STRIDE + (c2 << 1));
    }
#else
    float2 t[4];
    #pragma unroll
    for (int j = 0; j < 4; ++j) {
        int i  = tid + j * 256;
        int r  = i >> 3;
        int c2 = i & 7;
        t[j] = *(const float2*)(gsrc + (size_t)r * DIM + (c2 << 1));
    }
    #pragma unroll
    for (int j = 0; j < 4; ++j) {
        int i  = tid + j * 256;
        int r  = i >> 3;
        int c2 = i & 7;
        buf[r * C_STRIDE + (c2 << 1)]     = t[j].x;
        buf[r * C_STRIDE + (c2 << 1) + 1] = t[j].y;
    }
#endif
}

__global__ __launch_bounds__(256) void ge2e_score(const float* __restrict__ e_n,
                                                  const float* __restrict__ c_n,
                                                  const float* __restrict__ wp,
                                                  const float* __restrict__ bp,
                                                  float* __restrict__ out) {
    __shared__ float E[N_UTTS * E_STRIDE];
    __shared__ float Cc0[128 * C_STRIDE];
    __shared__ float Cc1[128 * C_STRIDE];
    __shared__ float S[16 * S_STRIDE];
    __shared__ float blockred[16];

    const int st  = blockIdx.x;    // s-tile (0..63)
    const int u   = blockIdx.y;    // utterance (0..15)
    const int tid = threadIdx.x;
    const int lane = tid & 31;
    const int wv   = tid >> 5;     // wave id 0..7

    const float* Ebase = e_n + ((size_t)u * N_SPKS + st * 16) * DIM;
    const float* Cbase = c_n + (size_t)u * N_SPKS * DIM;

    // Stage the E panel (16 x 512 floats) -- async when available.
#if GE2E_ASYNC_LDS
    #pragma unroll
    for (int j = 0; j < 16; ++j) {
        int i  = tid + j * 256;   // float2 index over 4096 (256 per row)
        int r  = i >> 8;
        int c2 = i & 255;
        ASYNC_CP_B64(Ebase + (size_t)r * DIM + (c2 << 1),
                     E + r * E_STRIDE + (c2 << 1));
    }
#else
    {
        float2 t[16];
        #pragma unroll
        for (int j = 0; j < 16; ++j) t[j] = ((const float2*)Ebase)[tid + j * 256];
        #pragma unroll
        for (int j = 0; j < 16; ++j) {
            int i  = tid + j * 256;
            int r  = i >> 8;
            int c2 = (i & 255) << 1;
            E[r * E_STRIDE + c2]     = t[j].x;
            E[r * E_STRIDE + c2 + 1] = t[j].y;
        }
    }
#endif

    const float w_s = wp[0];
    const float b_s = bp[0];

    // Per-lane WMMA operand addressing (A and B share the same lane pattern):
    // lane L (L<16): row L, K elems {k, k+1}; lane L>=16: row L-16, {k+2, k+3}
    const int rA   = lane & 15;
    const int koff = (lane >> 4) << 1;
    const float* eptr  = &E[rA * E_STRIDE + koff];
    const float* cp0   = &Cc0[(wv * 16 + rA) * C_STRIDE + koff];
    const float* cp1   = &Cc1[(wv * 16 + rA) * C_STRIDE + koff];

    // Online logsumexp state, held by threads 0..15 (one per s-row of tile).
    float m = -3.0e38f, z = 0.0f, diag = 0.0f;
    const int sglob = st * 16 + tid;   // meaningful for tid < 16

    for (int tc = 0; tc < 8; ++tc) {            // t-chunks of 128 columns
        const float* Ct = Cbase + (size_t)(tc * 128) * DIM;
        v8f acc = {0.f, 0.f, 0.f, 0.f, 0.f, 0.f, 0.f, 0.f};

        ge2e_stage_chunk(Ct, Cc0, tid);         // prologue: chunk 0 -> buf0

        #pragma unroll 4
        for (int kc = 0; kc < (DIM / KC); kc += 2) {   // 32 chunks, 2 per iter
            // ---- even chunk: consume Cc0, prefetch kc+1 into Cc1 ----
            ge2e_wait_async();
            __syncthreads();
            if (kc + 1 < (DIM / KC)) ge2e_stage_chunk(Ct + (kc + 1) * KC, Cc1, tid);
            {
                const float* ep = eptr + kc * KC;
                #pragma unroll
                for (int kk = 0; kk < KC / 4; ++kk) {
                    v2f a = *(const v2f*)(ep  + kk * 4);
                    v2f b = *(const v2f*)(cp0 + kk * 4);
                    acc = __builtin_amdgcn_wmma_f32_16x16x4_f32(
                            false, a, false, b, (short)0, acc, false, false);
                }
            }
            // ---- odd chunk: consume Cc1, prefetch kc+2 into Cc0 ----
            ge2e_wait_async();
            __syncthreads();
            if (kc + 2 < (DIM / KC)) ge2e_stage_chunk(Ct + (kc + 2) * KC, Cc0, tid);
            {
                const float* ep = eptr + (kc + 1) * KC;
                #pragma unroll
                for (int kk = 0; kk < KC / 4; ++kk) {
                    v2f a = *(const v2f*)(ep  + kk * 4);
                    v2f b = *(const v2f*)(cp1 + kk * 4);
                    acc = __builtin_amdgcn_wmma_f32_16x16x4_f32(
                            false, a, false, b, (short)0, acc, false, false);
                }
            }
        }

        // Spill the 16x16 tile: acc[r], lane L -> M = r + 8*(L>>4), N = L&15.
        __syncthreads();
        #pragma unroll
        for (int r = 0; r < 8; ++r) {
            int row = r + ((lane >> 4) << 3);
            int col = wv * 16 + (lane & 15);
            S[row * S_STRIDE + col] = acc[r];
        }
        __syncthreads();

        // Online logsumexp over this 128-column chunk; grab diagonal logit.
        if (tid < 16) {
            for (int c = 0; c < 128; ++c) {
                float v = w_s * S[tid * S_STRIDE + c] + b_s;
                int tg = tc * 128 + c;
                if (tg == sglob) diag = v;
                float mn = fmaxf(m, v);
                z = z * expf(m - mn) + expf(v - mn);
                m = mn;
            }
        }
        // Next tc's first wait+barrier separates these S reads from S rewrites.
    }

    // contrib_s = logsumexp_t(logits) - logits[s,u,s]
    if (tid < 16) blockred[tid] = (m + logf(z)) - diag;
    __syncthreads();
    if (tid == 0) {
        float sum = 0.0f;
        #pragma unroll
        for (int i = 0; i < 16; ++i) sum += blockred[i];
        atomicAdd(out, sum);
    }
}

// ---------------------------------------------------------------------------
extern "C" void kernel_launch(void* const* d_in, const int* in_sizes, int n_in,
                              void* d_out, int out_size, void* d_ws, size_t ws_size,
                              hipStream_t stream) {
    const float* emb = (const float*)d_in[0];   // (1024, 16, 512) fp32
    const float* w   = (const float*)d_in[1];   // (1,1)
    const float* b   = (const float*)d_in[2];   // (1,1)
    float* out = (float*)d_out;                 // scalar loss

    float* e_n = (float*)d_ws;                                   // 32 MB
    float* c_n = e_n + (size_t)N_UTTS * N_SPKS * DIM;            // 32 MB

    ge2e_zero_out<<<1, 64, 0, stream>>>(out);
    ge2e_prep<<<N_SPKS, 256, 0, stream>>>(emb, e_n, c_n);

    dim3 grid(N_SPKS / 16, N_UTTS);   // (64 s-tiles, 16 u)
    ge2e_score<<<grid, 256, 0, stream>>>(e_n, c_n, w, b, out);
}